// CtcLoss_3650722201614
// MI455X (gfx1250) — compile-verified
//
#include <hip/hip_runtime.h>
#include <stdint.h>

#define NEGV (-1e30f)

// ---------------------------------------------------------------------------
// CDNA5 async global->LDS helpers (gfx1250). One wave (32 lanes) pulls a
// 512-byte row (V=128 floats) into LDS with a single b128 async op per lane.
// Tracked with ASYNCcnt; consumer waits with s_wait_asynccnt then barriers.
// ---------------------------------------------------------------------------
__device__ __forceinline__ void async_row_load_b128(float* lds_row,
                                                    const float* gsrc,
                                                    int lane) {
    // Low 32 bits of a generic pointer derived from LDS are the LDS byte
    // offset.
    unsigned lds_addr = (unsigned)(size_t)(lds_row + lane * 4);
    unsigned long long gaddr = (unsigned long long)(size_t)(gsrc + lane * 4);
    asm volatile("global_load_async_to_lds_b128 %0, %1, off"
                 :: "v"(lds_addr), "v"(gaddr)
                 : "memory");
}

__device__ __forceinline__ void wait_async0() {
#if __has_builtin(__builtin_amdgcn_s_wait_asynccnt)
    __builtin_amdgcn_s_wait_asynccnt(0);
#else
    asm volatile("s_wait_asynccnt 0x0" ::: "memory");
#endif
}

// ---------------------------------------------------------------------------
// Kernel 1: lseT[b*T + t] = logsumexp_v logits[t,b,v].  One wave32 per (t,b)
// row; each lane owns one float4 (V=128 -> single b128 per lane, values kept
// in registers for both the max and the sum pass).  Output is TRANSPOSED so
// the scan kernel's per-step uniform load walks a contiguous 4B stream.
// ---------------------------------------------------------------------------
__global__ __launch_bounds__(256) void ctc_lse_kernel(
    const float* __restrict__ logits, float* __restrict__ lseT,
    int T, int B, int V) {
    int gwave = (blockIdx.x * blockDim.x + threadIdx.x) >> 5;
    int lane  = threadIdx.x & 31;
    if (gwave >= T * B) return;
    const float* row = logits + (size_t)gwave * V;

    float m, s;
    if (V == 128) {
        float4 x = ((const float4*)row)[lane];      // one b128 per lane
        m = fmaxf(fmaxf(x.x, x.y), fmaxf(x.z, x.w));
        for (int off = 16; off > 0; off >>= 1) m = fmaxf(m, __shfl_xor(m, off, 32));
        s = __expf(x.x - m) + __expf(x.y - m) + __expf(x.z - m) + __expf(x.w - m);
        for (int off = 16; off > 0; off >>= 1) s += __shfl_xor(s, off, 32);
    } else {
        m = NEGV;
        for (int v = lane; v < V; v += 32) m = fmaxf(m, row[v]);
        for (int off = 16; off > 0; off >>= 1) m = fmaxf(m, __shfl_xor(m, off, 32));
        s = 0.f;
        for (int v = lane; v < V; v += 32) s += __expf(row[v] - m);
        for (int off = 16; off > 0; off >>= 1) s += __shfl_xor(s, off, 32);
    }
    if (lane == 0) {
        int t = gwave / B, b = gwave - t * B;
        lseT[(size_t)b * T + t] = m + __logf(s);
    }
}

// ---------------------------------------------------------------------------
// Kernel 2: forward CTC scan. One block per batch element b.
// LDS: 2 row buffers (double-buffered via async DMA), 2 alpha buffers
// (padded with 2 leading NEG cells), ext/allow table.  One barrier per step.
// ---------------------------------------------------------------------------
__global__ __launch_bounds__(544) void ctc_alpha_kernel(
    const float* __restrict__ logits, const int* __restrict__ target,
    const int* __restrict__ in_len, const int* __restrict__ tgt_len,
    const float* __restrict__ lseT, float* __restrict__ negll,
    int T, int B, int V, int L) {
    const int b   = blockIdx.x;
    const int tid = threadIdx.x;
    const int S   = 2 * L + 1;

    extern __shared__ float smem[];
    float* row0 = smem;                 // V
    float* row1 = row0 + V;             // V
    float* a0   = row1 + V;             // S + 2
    float* a1   = a0 + (S + 2);         // S + 2
    int*   ea   = (int*)(a1 + (S + 2)); // S: ext | (allow<<16)

    const int len = in_len[b];
    const int tl  = tgt_len[b];
    const float* lseb = lseT + (size_t)b * T;   // contiguous per-b lse stream

    // ext / allow tables
    for (int s = tid; s < S; s += blockDim.x) {
        int e = 0, al = 0;
        if (s & 1) {
            int k = s >> 1;
            e = target[b * L + k];
            if (s >= 3) al = (target[b * L + k] != target[b * L + k - 1]) ? 1 : 0;
        }
        ea[s] = (e & 0xffff) | (al << 16);
    }
    // alpha pads (never overwritten afterwards)
    if (tid < 2) { a0[tid] = NEGV; a1[tid] = NEGV; }
    // alpha @ t=0
    if (tid < S) {
        float v = NEGV;
        if (tid == 0) {
            v = logits[(size_t)b * V + 0] - lseb[0];
        } else if (tid == 1) {
            v = logits[(size_t)b * V + target[b * L + 0]] - lseb[0];
        }
        a0[2 + tid] = v;
    }
    __syncthreads();

    // prefetch row for t=1 into buffer (1 & 1) = row1
    if (len > 1 && tid < 32)
        async_row_load_b128(row1, logits + ((size_t)1 * B + b) * V, tid);

    for (int t = 1; t < len; ++t) {
        float*       rcur  = (t & 1) ? row1 : row0;
        float*       rnext = (t & 1) ? row0 : row1;
        const float* aprev = ((t - 1) & 1) ? a1 : a0;
        float*       acur  = (t & 1) ? a1 : a0;

        if (tid < 32) wait_async0();   // row t landed in LDS
        __syncthreads();               // visible to all; prev compute done

        if (t + 1 < len && tid < 32)   // overlap next row DMA with compute
            async_row_load_b128(rnext, logits + ((size_t)(t + 1) * B + b) * V, tid);

        // keep the lse stream ahead of the scan (global_prefetch_b8)
        if ((t & 31) == 0 && t + 160 < T && tid == 0)
            __builtin_prefetch(lseb + t + 160, 0, 3);

        float lset = lseb[t];          // uniform, contiguous -> scalar load
        if (tid < S) {
            int   e  = ea[tid];
            float a  = aprev[2 + tid];
            float p1 = aprev[1 + tid];
            float p2 = (e & 0x10000) ? aprev[tid] : NEGV;
            float m  = fmaxf(a, fmaxf(p1, p2));
            float sm = __expf(a - m) + __expf(p1 - m) + __expf(p2 - m);
            acur[2 + tid] = m + __logf(sm) + (rcur[e & 0xffff] - lset);
        }
        // no trailing barrier: next iteration's barrier orders acur reads
    }
    __syncthreads();

    if (tid == 0) {
        const float* A = ((len - 1) & 1) ? a1 : a0;
        float la = A[2 + 2 * tl];
        float lb = A[1 + 2 * tl];
        float m  = fmaxf(la, lb);
        float ll = m + __logf(__expf(la - m) + __expf(lb - m));
        negll[b] = -ll;
    }
}

// ---------------------------------------------------------------------------
// Kernel 3: deterministic fixed-order reduction of per-b losses -> scalar.
// ---------------------------------------------------------------------------
__global__ __launch_bounds__(256) void ctc_reduce_kernel(
    const float* __restrict__ negll, float* __restrict__ out, int B) {
    __shared__ float sh[256];
    int i = threadIdx.x;
    float v = 0.f;
    for (int j = i; j < B; j += 256) v += negll[j];
    sh[i] = v;
    __syncthreads();
    for (int off = 128; off > 0; off >>= 1) {
        if (i < off) sh[i] += sh[i + off];
        __syncthreads();
    }
    if (i == 0) out[0] = sh[0];
}

// ---------------------------------------------------------------------------
extern "C" void kernel_launch(void* const* d_in, const int* in_sizes, int n_in,
                              void* d_out, int out_size, void* d_ws, size_t ws_size,
                              hipStream_t stream) {
    const float* logits  = (const float*)d_in[0];
    const int*   target  = (const int*)d_in[1];
    const int*   in_len  = (const int*)d_in[2];
    const int*   tgt_len = (const int*)d_in[3];

    const int B = in_sizes[2];           // 64
    const int L = in_sizes[1] / B;       // 256
    const int V = 128;                   // vocab (fixed by reference)
    const int T = in_sizes[0] / (B * V); // 2000
    const int S = 2 * L + 1;             // 513

    float* lseT  = (float*)d_ws;               // B*T floats (transposed)
    float* negll = lseT + (size_t)T * B;       // B floats

    // Pass 1: logsumexp rows (one wave32 per (t,b))
    int rows = T * B;
    int wavesPerBlock = 256 / 32;
    dim3 g1((rows + wavesPerBlock - 1) / wavesPerBlock), b1(256);
    ctc_lse_kernel<<<g1, b1, 0, stream>>>(logits, lseT, T, B, V);

    // Pass 2: per-batch forward scan with async LDS double-buffering
    int bt = ((S + 31) / 32) * 32;             // 544 threads
    size_t smem = (size_t)(2 * V + 2 * (S + 2) + S) * sizeof(float);
    ctc_alpha_kernel<<<B, bt, smem, stream>>>(logits, target, in_len, tgt_len,
                                              lseT, negll, T, B, V, L);

    // Pass 3: deterministic scalar reduction
    ctc_reduce_kernel<<<1, 256, 0, stream>>>(negll, (float*)d_out, B);
}